// GCN_463856468204
// MI455X (gfx1250) — compile-verified
//
#include <hip/hip_runtime.h>

#define NN 50000
#define NE 800000
#define D  64
#define NBLK ((NN + 255) / 256)   // 196 scan blocks

typedef __attribute__((ext_vector_type(2))) float v2f;
typedef __attribute__((ext_vector_type(8))) float v8f;
typedef __attribute__((ext_vector_type(4))) unsigned int v4u;
typedef __attribute__((ext_vector_type(8))) int v8i_;
typedef __attribute__((ext_vector_type(4))) int v4i_;

#if defined(__has_builtin)
#if __has_builtin(__builtin_amdgcn_tensor_load_to_lds) && __has_builtin(__builtin_amdgcn_s_wait_tensorcnt)
#define USE_TDM 1
#endif
#endif

// ------------------------------------------------------------- init/degrees -
__global__ void k_init(int* cs, int* cd, int* cursor) {
    int i = blockIdx.x * 256 + threadIdx.x;
    if (i < NN) { cs[i] = 0; cd[i] = 0; cursor[i] = 0; }
}

__global__ void k_count(const int* __restrict__ ei, int* cs, int* cd) {
    int e = blockIdx.x * 256 + threadIdx.x;
    if (e < NE) {
        atomicAdd(&cs[ei[e]],      1);
        atomicAdd(&cd[ei[NE + e]], 1);
    }
}

__global__ void k_norm(const int* __restrict__ cs, const int* __restrict__ cd,
                       float* ns, float* ndst) {
    int i = blockIdx.x * 256 + threadIdx.x;
    if (i < NN) {                       // +1 = self loop; degree always >= 1
        ns[i]   = rsqrtf((float)(cs[i] + 1));
        ndst[i] = rsqrtf((float)(cd[i] + 1));
    }
}

// ------------------------------------------------- exclusive scan (3 phases) -
__global__ void k_scan_block(const int* __restrict__ cnt, int* __restrict__ rowstart,
                             int* __restrict__ bsum) {
    __shared__ int sh[256];
    int i = blockIdx.x * 256 + threadIdx.x;
    int v = (i < NN) ? cnt[i] : 0;
    sh[threadIdx.x] = v;
    __syncthreads();
#pragma unroll
    for (int off = 1; off < 256; off <<= 1) {
        int t = (threadIdx.x >= off) ? sh[threadIdx.x - off] : 0;
        __syncthreads();
        sh[threadIdx.x] += t;
        __syncthreads();
    }
    if (i < NN) rowstart[i] = sh[threadIdx.x] - v;        // exclusive within block
    if (threadIdx.x == 255) bsum[blockIdx.x] = sh[255];
}

__global__ void k_scan_sums(int* bsum) {                  // single block
    __shared__ int sh[256];
    int v = (threadIdx.x < NBLK) ? bsum[threadIdx.x] : 0;
    sh[threadIdx.x] = v;
    __syncthreads();
#pragma unroll
    for (int off = 1; off < 256; off <<= 1) {
        int t = (threadIdx.x >= off) ? sh[threadIdx.x - off] : 0;
        __syncthreads();
        sh[threadIdx.x] += t;
        __syncthreads();
    }
    if (threadIdx.x < NBLK) bsum[threadIdx.x] = sh[threadIdx.x] - v;   // exclusive
}

__global__ void k_scan_add(int* rowstart, const int* __restrict__ bsum) {
    int i = blockIdx.x * 256 + threadIdx.x;
    if (i < NN) rowstart[i] += bsum[i >> 8];
}

// -------------------------------------------------------------- CSR fill -----
__global__ void k_fill(const int* __restrict__ ei, const int* __restrict__ rowstart,
                       int* cursor, int* __restrict__ csr) {
    int e = blockIdx.x * 256 + threadIdx.x;
    if (e < NE) {
        int s = ei[e], d = ei[NE + e];
        int pos = atomicAdd(&cursor[d], 1);
        csr[rowstart[d] + pos] = s;
    }
}

// -------------------------- pull aggregation: wave32 per dst node, no atomics
__global__ void __launch_bounds__(256)
k_agg_pull(const float* __restrict__ x, const float* __restrict__ ns,
           const int* __restrict__ rowstart, const int* __restrict__ cnt,
           const int* __restrict__ csr, float* __restrict__ agg) {
    int tid  = blockIdx.x * 256 + threadIdx.x;
    int node = tid >> 5;                 // one wave per destination node
    int lane = tid & 31;                 // lane owns columns 2l, 2l+1
    if (node >= NN) return;
    const float2* xv = (const float2*)x;
    float2 a = xv[(long)node * 32 + lane];          // self-loop term
    float  w = ns[node];
    float2 acc; acc.x = a.x * w; acc.y = a.y * w;
    int rs = rowstart[node], rc = cnt[node];
    for (int j = 0; j < rc; j++) {                  // rs, rc, s wave-uniform
        int s   = csr[rs + j];
        float v = ns[s];
        float2 xr = xv[(long)s * 32 + lane];        // coalesced 256B per wave
        acc.x += xr.x * v;
        acc.y += xr.y * v;
    }
    ((float2*)agg)[(long)node * 32 + lane] = acc;
}

// -------------------------------------------- dense GEMM + bias + ReLU ------
// h[n,:] = relu( (agg[n,:] * norm_dst[n]) @ W1 + b1 ),  V_WMMA_F32_16X16X4_F32
__global__ void __launch_bounds__(256)
k_gemm_relu(const float* __restrict__ agg, const float* __restrict__ nd,
            const float* __restrict__ W1,  const float* __restrict__ b1,
            float* __restrict__ h) {
    __shared__ float ldsW[D * D];   // 16 KB W1 tile shared by 8 waves
    __shared__ float ldsB[D];

    int tid = threadIdx.x;
#ifdef USE_TDM
    if (tid < 32) {                  // wave 0 issues one TDM copy of all of W1
        unsigned long long ga = (unsigned long long)(size_t)W1;
        unsigned lds_addr = (unsigned)(size_t)ldsW;     // generic LDS ptr low 32b = LDS offset
        v4u g0;
        g0[0] = 1u;                                     // count=1, load, no gather
        g0[1] = lds_addr;                               // lds_addr [63:32]
        g0[2] = (unsigned)(ga & 0xffffffffu);           // global_addr [95:64]
        g0[3] = (unsigned)((ga >> 32) & 0x01ffffffu) | (2u << 30);  // addr hi + type=2
        v8i_ g1;
        g1[0] = 0x20000;            // workgroup_mask=0, data_size=2 (4B)
        g1[1] = (int)(4096u << 16); // tensor_dim0[15:0]=4096 at bits[63:48]
        g1[2] = (int)(1u << 16);    // tensor_dim0 hi=0, tensor_dim1[15:0]=1
        g1[3] = (int)(4096u << 16); // tensor_dim1 hi=0, tile_dim0=4096
        g1[4] = 1;                  // tile_dim1=1, tile_dim2=0
        g1[5] = 4096;               // tensor_dim0_stride lo
        g1[6] = 0;                  // stride hi, dim1_stride lo
        g1[7] = 0;
        v4i_ gz4 = {0, 0, 0, 0};
        v8i_ gz8 = {0, 0, 0, 0, 0, 0, 0, 0};
        // amdgpu-toolchain (clang-23) 6-arg form: (g0, g1, g2, g3, g?, cpol)
        __builtin_amdgcn_tensor_load_to_lds(g0, g1, gz4, gz4, gz8, 0);
        __builtin_amdgcn_s_wait_tensorcnt(0);
    }
#else
    {
        const float4* w4 = (const float4*)W1;
        float4*       l4 = (float4*)ldsW;
#pragma unroll
        for (int i = 0; i < 4; i++) l4[i * 256 + tid] = w4[i * 256 + tid];
    }
#endif
    if (tid < D) ldsB[tid] = b1[tid];
    __syncthreads();

    int wave = tid >> 5, lane = tid & 31;
    int rowTile = blockIdx.x * 8 + wave;        // 50000/16 = 3125 row tiles
    if (rowTile >= NN / 16) return;             // wave-uniform: EXEC all-ones
    int row0 = rowTile * 16;

    // A 16x4 f32 layout: lanes 0-15 -> M=lane, K={0,1}; lanes 16-31 -> K={2,3}
    int m  = lane & 15;
    int kb = (lane >> 4) << 1;
    float ndv = nd[row0 + m];
    const float* arow = agg + (long)(row0 + m) * D;

    v2f afrag[16];
#pragma unroll
    for (int ks = 0; ks < 16; ks++) {
        afrag[ks].x = arow[ks * 4 + kb]     * ndv;
        afrag[ks].y = arow[ks * 4 + kb + 1] * ndv;
    }

#pragma unroll
    for (int ct = 0; ct < 4; ct++) {            // 4 col tiles -> full D=64
        int n = ct * 16 + (lane & 15);
        v8f c = {};
#pragma unroll
        for (int ks = 0; ks < 16; ks++) {       // K=64 in steps of 4
            int kk = ks * 4 + kb;
            v2f b;
            b.x = ldsW[kk * D + n];
            b.y = ldsW[(kk + 1) * D + n];
            c = __builtin_amdgcn_wmma_f32_16x16x4_f32(
                    false, afrag[ks], false, b, (short)0, c, false, false);
        }
        float bb = ldsB[n];
#pragma unroll
        for (int r = 0; r < 8; r++) {           // C/D: VGPR r -> M=r / r+8
            int M = r + ((lane >> 4) << 3);
            float v = c[r] + bb;
            h[(long)(row0 + M) * D + n] = v > 0.0f ? v : 0.0f;
        }
    }
}

// ------------------------------- layer 2: matvec (pre-scaled) + pull + bias -
__global__ void k_matvec(const float* __restrict__ h, const float* __restrict__ W2,
                         const float* __restrict__ ns, float* __restrict__ z) {
    int tid  = blockIdx.x * 256 + threadIdx.x;
    int row  = tid >> 5;                        // one wave32 per row
    int lane = tid & 31;
    if (row >= NN) return;
    const float* hr = h + (long)row * D;
    float s = hr[lane] * W2[lane] + hr[lane + 32] * W2[lane + 32];
#pragma unroll
    for (int off = 16; off > 0; off >>= 1) s += __shfl_down(s, off, 32);
    if (lane == 0) z[row] = s * ns[row];        // pre-scale by norm_src
}

__global__ void k_out_pull(const int* __restrict__ rowstart, const int* __restrict__ cnt,
                           const int* __restrict__ csr, const float* __restrict__ z,
                           const float* __restrict__ ndst, const float* __restrict__ b2,
                           float* __restrict__ out) {
    int n = blockIdx.x * 256 + threadIdx.x;
    if (n >= NN) return;
    float s = z[n];                             // self loop
    int rs = rowstart[n], rc = cnt[n];
    for (int j = 0; j < rc; j++) s += z[csr[rs + j]];
    out[n] = s * ndst[n] + b2[0];
}

// ---------------------------------------------------------------------------
extern "C" void kernel_launch(void* const* d_in, const int* in_sizes, int n_in,
                              void* d_out, int out_size, void* d_ws, size_t ws_size,
                              hipStream_t stream) {
    const float* x  = (const float*)d_in[0];
    const int*   ei = (const int*)  d_in[1];
    const float* W1 = (const float*)d_in[2];
    const float* b1 = (const float*)d_in[3];
    const float* W2 = (const float*)d_in[4];
    const float* b2 = (const float*)d_in[5];
    float* out = (float*)d_out;

    int*   cs       = (int*)d_ws;          // N  out-degree counts (no self loop)
    int*   cd       = cs + NN;             // N  in-degree counts
    int*   cursor   = cd + NN;             // N  CSR fill cursors
    int*   rowstart = cursor + NN;         // N  exclusive scan of cd
    int*   bsum     = rowstart + NN;       // 256 block sums
    int*   csr      = bsum + 256;          // NE src ids grouped by dst
    float* ns       = (float*)(csr + NE);  // N  norm_src
    float* ndst     = ns + NN;             // N  norm_dst
    float* agg      = ndst + NN;           // N*64
    float* hbuf     = agg + NN * D;        // N*64
    float* zbuf     = hbuf + NN * D;       // N      (~30 MB total)

    const int nb_n  = NBLK;                // 196
    const int nb_e  = (NE + 255) / 256;    // 3125
    const int nb_w  = (NN * 32) / 256;     // 6250 (wave per node)
    const int nb_g  = (NN / 16 + 7) / 8;   // 391

    k_init      <<<nb_n, 256, 0, stream>>>(cs, cd, cursor);
    k_count     <<<nb_e, 256, 0, stream>>>(ei, cs, cd);
    k_norm      <<<nb_n, 256, 0, stream>>>(cs, cd, ns, ndst);
    k_scan_block<<<nb_n, 256, 0, stream>>>(cd, rowstart, bsum);
    k_scan_sums <<<1,    256, 0, stream>>>(bsum);
    k_scan_add  <<<nb_n, 256, 0, stream>>>(rowstart, bsum);
    k_fill      <<<nb_e, 256, 0, stream>>>(ei, rowstart, cursor, csr);
    k_agg_pull  <<<nb_w, 256, 0, stream>>>(x, ns, rowstart, cd, csr, agg);
    k_gemm_relu <<<nb_g, 256, 0, stream>>>(agg, ndst, W1, b1, hbuf);
    k_matvec    <<<nb_w, 256, 0, stream>>>(hbuf, W2, ns, zbuf);
    k_out_pull  <<<nb_n, 256, 0, stream>>>(rowstart, cd, csr, zbuf, ndst, b2, out);
}